// GNNConv_42880953483584
// MI455X (gfx1250) — compile-verified
//
#include <hip/hip_runtime.h>
#include <hip/hip_bf16.h>

typedef __attribute__((ext_vector_type(2))) float v2f;
typedef __attribute__((ext_vector_type(8))) float v8f;

#define DDIM 64
#define NREL 8
#define KTOT 512  // NREL * DDIM

// ---------------- zero workspace ----------------
__global__ void zero_ws_kernel(float* __restrict__ ws, long n) {
    long i = (long)blockIdx.x * blockDim.x + threadIdx.x;
    long stride = (long)gridDim.x * blockDim.x;
    float4 z = {0.f, 0.f, 0.f, 0.f};
    float4* p = (float4*)ws;
    long n4 = n >> 2;
    for (long j = i; j < n4; j += stride) p[j] = z;
    if (i == 0) for (long j = n4 << 2; j < n; ++j) ws[j] = 0.f;
}

// ---------------- per-(dst,rel) in-degree counts ----------------
__global__ void count_kernel(const int* __restrict__ ei, const int* __restrict__ et,
                             float* __restrict__ cnt, int E) {
    int e = blockIdx.x * blockDim.x + threadIdx.x;
    if (e >= E) return;
    int dst = ei[(long)E + e];
    int r   = et[e];
    atomicAdd(&cnt[(long)dst * NREL + r], 1.0f);
}

// ---------------- scatter pre-normalized source rows: one wave per edge ----------------
__global__ __launch_bounds__(256)
void scatter_kernel(const float* __restrict__ x, const int* __restrict__ ei,
                    const int* __restrict__ et, const float* __restrict__ cnt,
                    float* __restrict__ s, int E) {
    int wave = (int)(((long)blockIdx.x * blockDim.x + threadIdx.x) >> 5);
    int lane = threadIdx.x & 31;
    if (wave >= E) return;
    // wave-uniform edge scalars -> scalar loads
    int src = ei[wave];
    int dst = ei[(long)E + wave];
    int r   = et[wave];
    float c   = cnt[(long)dst * NREL + r];
    float inv = 1.0f / fmaxf(c, 1.0f);
    const float2* xp = (const float2*)(x + (long)src * DDIM);
    float* sp = s + ((long)dst * NREL + r) * DDIM + lane * 2;
    float2 v = xp[lane];
    atomicAdd(&sp[0], v.x * inv);
    atomicAdd(&sp[1], v.y * inv);
}

// ---------------- fused GEMM: out = [s | x] @ [W_rel ; W_root] + bias ----------------
// One wave computes a 16-row x 64-col tile with 4 f32 WMMA accumulators.
__global__ __launch_bounds__(256)
void rgcn_gemm_kernel(const float* __restrict__ s, const float* __restrict__ x,
                      const float* __restrict__ Wrel, const float* __restrict__ Wroot,
                      const float* __restrict__ bias, float* __restrict__ out, int ntiles) {
    int tile = blockIdx.x * 8 + (threadIdx.x >> 5);   // wave-uniform
    if (tile >= ntiles) return;
    int lane = threadIdx.x & 31;
    int row  = lane & 15;    // M within tile (A) / column within 16 (B,C)
    int kh   = lane >> 4;    // 0: K pair {0,1}; 1: K pair {2,3}
    long n0 = (long)tile * 16;

    const float* srow = s + (n0 + row) * (long)KTOT + 2 * kh;
    const float* xrow = x + (n0 + row) * (long)DDIM + 2 * kh;

    v8f c0 = {}, c1 = {}, c2 = {}, c3 = {};

    // Phase 1: K = 0..511 over normalized relation sums, B = stacked W_rel
#pragma unroll 4
    for (int kk = 0; kk < KTOT; kk += 4) {
        if ((kk & 63) == 0) __builtin_prefetch(srow + kk + 64, 0, 1);
        float2 av = *(const float2*)(srow + kk);
        v2f a; a.x = av.x; a.y = av.y;
        const float* wk = Wrel + (long)(kk + 2 * kh) * DDIM + row;
        v2f b0, b1, b2, b3;
        b0.x = wk[0];  b0.y = wk[DDIM + 0];
        b1.x = wk[16]; b1.y = wk[DDIM + 16];
        b2.x = wk[32]; b2.y = wk[DDIM + 32];
        b3.x = wk[48]; b3.y = wk[DDIM + 48];
        c0 = __builtin_amdgcn_wmma_f32_16x16x4_f32(false, a, false, b0, (short)0, c0, false, false);
        c1 = __builtin_amdgcn_wmma_f32_16x16x4_f32(false, a, false, b1, (short)0, c1, false, false);
        c2 = __builtin_amdgcn_wmma_f32_16x16x4_f32(false, a, false, b2, (short)0, c2, false, false);
        c3 = __builtin_amdgcn_wmma_f32_16x16x4_f32(false, a, false, b3, (short)0, c3, false, false);
    }

    // Phase 2: K = 0..63 over x, B = W_root (root/self transform)
#pragma unroll 4
    for (int kk = 0; kk < DDIM; kk += 4) {
        float2 av = *(const float2*)(xrow + kk);
        v2f a; a.x = av.x; a.y = av.y;
        const float* wk = Wroot + (long)(kk + 2 * kh) * DDIM + row;
        v2f b0, b1, b2, b3;
        b0.x = wk[0];  b0.y = wk[DDIM + 0];
        b1.x = wk[16]; b1.y = wk[DDIM + 16];
        b2.x = wk[32]; b2.y = wk[DDIM + 32];
        b3.x = wk[48]; b3.y = wk[DDIM + 48];
        c0 = __builtin_amdgcn_wmma_f32_16x16x4_f32(false, a, false, b0, (short)0, c0, false, false);
        c1 = __builtin_amdgcn_wmma_f32_16x16x4_f32(false, a, false, b1, (short)0, c1, false, false);
        c2 = __builtin_amdgcn_wmma_f32_16x16x4_f32(false, a, false, b2, (short)0, c2, false, false);
        c3 = __builtin_amdgcn_wmma_f32_16x16x4_f32(false, a, false, b3, (short)0, c3, false, false);
    }

    // Epilogue: C layout -> VGPR v holds M=v (lanes 0-15) / M=v+8 (lanes 16-31), col = lane&15
    float bc0 = bias[row], bc1 = bias[16 + row], bc2 = bias[32 + row], bc3 = bias[48 + row];
#pragma unroll
    for (int v = 0; v < 8; ++v) {
        long m = n0 + v + kh * 8;
        float* o = out + m * DDIM + row;
        o[0]  = c0[v] + bc0;
        o[16] = c1[v] + bc1;
        o[32] = c2[v] + bc2;
        o[48] = c3[v] + bc3;
    }
}

// ---------------- pass-through tuple outputs (edge_index, edge_type) ----------------
__global__ void copy_edges_kernel(const int* __restrict__ ei, const int* __restrict__ et,
                                  float* __restrict__ out_tail, int E) {
    long i = (long)blockIdx.x * blockDim.x + threadIdx.x;
    long two_e = 2L * E;
    if (i >= two_e + E) return;
    int v = (i < two_e) ? ei[i] : et[i - two_e];
    out_tail[i] = (float)v;
}

extern "C" void kernel_launch(void* const* d_in, const int* in_sizes, int n_in,
                              void* d_out, int out_size, void* d_ws, size_t ws_size,
                              hipStream_t stream) {
    const float* x     = (const float*)d_in[0];
    const int*   ei    = (const int*)d_in[1];   // [2, E] src then dst
    const int*   et    = (const int*)d_in[2];   // [E]
    const float* Wrel  = (const float*)d_in[3]; // [R, D, D]
    const float* Wroot = (const float*)d_in[4]; // [D, D]
    const float* bias  = (const float*)d_in[5]; // [D]
    float* out = (float*)d_out;

    int E = in_sizes[2];
    int N = in_sizes[0] / DDIM;

    // workspace: cnt [N*R] floats, then s [N*R*D] floats (~208 MB)
    float* cnt = (float*)d_ws;
    float* s   = cnt + (long)N * NREL;
    long zn = (long)N * NREL * (DDIM + 1);

    zero_ws_kernel<<<2048, 256, 0, stream>>>(cnt, zn);

    count_kernel<<<(E + 255) / 256, 256, 0, stream>>>(ei, et, cnt, E);

    long scatter_threads = (long)E * 32;
    scatter_kernel<<<(unsigned)((scatter_threads + 255) / 256), 256, 0, stream>>>(
        x, ei, et, cnt, s, E);

    int ntiles = (N + 15) / 16;
    rgcn_gemm_kernel<<<(ntiles + 7) / 8, 256, 0, stream>>>(
        s, x, Wrel, Wroot, bias, out, ntiles);

    long tail = 3L * E;
    copy_edges_kernel<<<(unsigned)((tail + 255) / 256), 256, 0, stream>>>(
        ei, et, out + (long)N * DDIM, E);
}